// DilatedAttention_73555609911682
// MI455X (gfx1250) — compile-verified
//
#include <hip/hip_runtime.h>
#include <hip/hip_bf16.h>
#include <math.h>

// ---------------------------------------------------------------------------
// Problem constants (from the reference)
// ---------------------------------------------------------------------------
#define BB   2
#define LL   2048
#define DD   512
#define HH   8
#define HD   64
#define DIL  2
#define WIN  256
#define HIDDEN 2048
#define NTOK (BB * LL)        // 4096 rows
#define EPSV 1e-5f

#define PADW 36               // LDS row stride (floats): 16B-aligned rows,
                              // row*36 mod 64 covers all bank groups

typedef float v2f __attribute__((ext_vector_type(2)));
typedef float v8f __attribute__((ext_vector_type(8)));

// ---------------------------------------------------------------------------
// LayerNorm: one 128-thread block per row of 512, float4 per thread
// ---------------------------------------------------------------------------
__global__ __launch_bounds__(128)
void ln_kernel(const float* __restrict__ x, const float* __restrict__ w,
               const float* __restrict__ b, float* __restrict__ out)
{
    const int row = blockIdx.x;
    const int t   = threadIdx.x;                 // 0..127
    const float4* xr = reinterpret_cast<const float4*>(x + (size_t)row * DD);
    float4 v = xr[t];

    float s  = v.x + v.y + v.z + v.w;
    float s2 = v.x * v.x + v.y * v.y + v.z * v.z + v.w * v.w;

    #pragma unroll
    for (int m = 16; m >= 1; m >>= 1) {
        s  += __shfl_xor(s,  m, 32);
        s2 += __shfl_xor(s2, m, 32);
    }
    __shared__ float ss[4], ss2[4];
    const int wid = t >> 5;
    if ((t & 31) == 0) { ss[wid] = s; ss2[wid] = s2; }
    __syncthreads();
    s  = ss[0]  + ss[1]  + ss[2]  + ss[3];
    s2 = ss2[0] + ss2[1] + ss2[2] + ss2[3];

    const float mean = s * (1.0f / DD);
    const float var  = s2 * (1.0f / DD) - mean * mean;
    const float rstd = rsqrtf(var + EPSV);

    const float4 wv = reinterpret_cast<const float4*>(w)[t];
    const float4 bv = reinterpret_cast<const float4*>(b)[t];
    float4 o;
    o.x = (v.x - mean) * rstd * wv.x + bv.x;
    o.y = (v.y - mean) * rstd * wv.y + bv.y;
    o.z = (v.z - mean) * rstd * wv.z + bv.z;
    o.w = (v.w - mean) * rstd * wv.w + bv.w;
    reinterpret_cast<float4*>(out + (size_t)row * DD)[t] = o;
}

// ---------------------------------------------------------------------------
// Tile epilogue: bias (+exact GELU) (+residual) and 16x16 store.
// ---------------------------------------------------------------------------
__device__ __forceinline__
void store_tile(v8f acc, int rbase, int col, int N,
                const float* __restrict__ bias,
                const float* __restrict__ residual,
                float* __restrict__ C, int apply_gelu)
{
    const float bval = bias ? bias[col] : 0.0f;
    #pragma unroll
    for (int r = 0; r < 8; ++r) {
        float v = acc[r] + bval;
        if (apply_gelu)
            v = 0.5f * v * (1.0f + erff(v * 0.70710678118654752f));  // exact GELU
        const size_t off = (size_t)(rbase + r) * N + col;
        if (residual) v += residual[off];
        C[off] = v;
    }
}

// ---------------------------------------------------------------------------
// Async-LDS stage of one 64-row x 32-K panel pair (A rows m0.., W rows n0..).
// 512 x 16B chunks per panel, 128 threads -> 4 chunks/thread/panel, via
// GLOBAL_LOAD_ASYNC_TO_LDS_B128 (ASYNCcnt).  Also prefetches the
// chunk-after-next toward L2 (global_prefetch_b8).
// ---------------------------------------------------------------------------
__device__ __forceinline__
void stage_panels(const float* __restrict__ A, const float* __restrict__ W,
                  int m0, int n0, int K, int kb,
                  float* sAbuf, float* sBbuf, int tid)
{
    #pragma unroll
    for (int i = 0; i < 4; ++i) {
        const int cid = tid + (i << 7);      // 0..511
        const int row = cid >> 3;            // 0..63
        const int c16 = cid & 7;             // 16B chunk within 128B row-slice

        const float* ga = A + (size_t)(m0 + row) * K + kb + (c16 << 2);
        const float* gb = W + (size_t)(n0 + row) * K + kb + (c16 << 2);
        const unsigned la =
            (unsigned)(unsigned long long)(sAbuf + row * PADW + (c16 << 2));
        const unsigned lb =
            (unsigned)(unsigned long long)(sBbuf + row * PADW + (c16 << 2));
        const unsigned long long gau = (unsigned long long)ga;
        const unsigned long long gbu = (unsigned long long)gb;

        asm volatile("global_load_async_to_lds_b128 %0, %1, off"
                     :: "v"(la), "v"(gau) : "memory");
        asm volatile("global_load_async_to_lds_b128 %0, %1, off"
                     :: "v"(lb), "v"(gbu) : "memory");
        if (c16 == 0) {                      // pull chunk-after-next into L2
            __builtin_prefetch(ga + 64, 0, 0);
            __builtin_prefetch(gb + 64, 0, 0);
        }
    }
}

// ---------------------------------------------------------------------------
// fp32 WMMA GEMM, 4 waves / workgroup, 64x64 C block per workgroup:
//   C[M,N] = A[M,K] @ W[N,K]^T (+ bias) (+ GELU) (+ residual)
// A/B K-panels double-buffered in LDS via async-LDS DMA; each wave computes a
// 32x32 sub-block (2x2 WMMA tiles) from LDS: per K-step of 4 -> 4 ds_load_b64
// + 4 V_WMMA_F32_16X16X4_F32.  Global traffic overlaps compute (ASYNCcnt).
//
// f32 fragment layouts (CDNA5 ISA 7.12.2):
//   A(16x4)/B(4x16): lanes 0-15 -> M/N=lane, K0..K1; lanes 16-31 -> K2..K3
//   C/D(16x16): lane&15 = column, 8 VGPR rows (+8 for lanes>=16)
// ---------------------------------------------------------------------------
__global__ __launch_bounds__(128)
void gemm_wmma_kernel(const float* __restrict__ A, const float* __restrict__ W,
                      const float* __restrict__ bias,
                      const float* __restrict__ residual,
                      float* __restrict__ C,
                      int M, int N, int K, int apply_gelu)
{
    __shared__ float sA[2][64 * PADW];
    __shared__ float sB[2][64 * PADW];

    const int n0   = blockIdx.x << 6;     // 64-wide N block
    const int m0   = blockIdx.y << 6;     // 64-tall M block
    const int tid  = threadIdx.x;         // 0..127
    const int lane = tid & 31;
    const int wave = tid >> 5;            // 0..3
    const int wm   = wave >> 1;           // wave's 32-row sub-block
    const int wn   = wave & 1;            // wave's 32-col sub-block
    const int half = lane >> 4;           // 0 or 1
    const int sub  = lane & 15;
    const int kk   = half << 1;           // lanes 0-15 -> K0,K1 ; 16-31 -> K2,K3

    // LDS fragment base pointers for this wave
    const float* aL0 = &sA[0][(32 * wm + sub) * PADW + kk];
    const float* aL1 = aL0 + 16 * PADW;
    const float* bL0 = &sB[0][(32 * wn + sub) * PADW + kk];
    const float* bL1 = bL0 + 16 * PADW;
    const int bufStride = 64 * PADW;      // floats between buffer 0 and 1

    v8f acc00 = {}, acc01 = {}, acc10 = {}, acc11 = {};

    // Prologue: stage chunk 0 into buffer 0
    stage_panels(A, W, m0, n0, K, 0, sA[0], sB[0], tid);
    asm volatile("s_wait_asynccnt 0x0" ::: "memory");
    __syncthreads();

    for (int kb = 0; kb < K; kb += 32) {
        const int cur = (kb >> 5) & 1;

        // Overlap: stage next chunk into the other buffer while computing
        if (kb + 32 < K)
            stage_panels(A, W, m0, n0, K, kb + 32,
                         sA[cur ^ 1], sB[cur ^ 1], tid);

        const float* a0p = aL0 + cur * bufStride;
        const float* a1p = aL1 + cur * bufStride;
        const float* b0p = bL0 + cur * bufStride;
        const float* b1p = bL1 + cur * bufStride;

        #pragma unroll
        for (int ku = 0; ku < 32; ku += 4) {
            const v2f a0 = *reinterpret_cast<const v2f*>(a0p + ku);
            const v2f a1 = *reinterpret_cast<const v2f*>(a1p + ku);
            const v2f b0 = *reinterpret_cast<const v2f*>(b0p + ku);
            const v2f b1 = *reinterpret_cast<const v2f*>(b1p + ku);

            acc00 = __builtin_amdgcn_wmma_f32_16x16x4_f32(
                        false, a0, false, b0, (short)0, acc00, false, false);
            acc01 = __builtin_amdgcn_wmma_f32_16x16x4_f32(
                        false, a0, false, b1, (short)0, acc01, false, false);
            acc10 = __builtin_amdgcn_wmma_f32_16x16x4_f32(
                        false, a1, false, b0, (short)0, acc10, false, false);
            acc11 = __builtin_amdgcn_wmma_f32_16x16x4_f32(
                        false, a1, false, b1, (short)0, acc11, false, false);
        }

        // Staged data LDS-visible before the barrier; buffer reuse safe after.
        asm volatile("s_wait_asynccnt 0x0" ::: "memory");
        __syncthreads();
    }

    const int rb0 = m0 + 32 * wm + (half << 3);
    const int rb1 = rb0 + 16;
    const int c0  = n0 + 32 * wn + sub;
    const int c1  = c0 + 16;

    store_tile(acc00, rb0, c0, N, bias, residual, C, apply_gelu);
    store_tile(acc01, rb0, c1, N, bias, residual, C, apply_gelu);
    store_tile(acc10, rb1, c0, N, bias, residual, C, apply_gelu);
    store_tile(acc11, rb1, c1, N, bias, residual, C, apply_gelu);
}

// ---------------------------------------------------------------------------
// Dilated windowed causal attention (sparse: <=129 keys/query).
// One wave per (b, h, q); lanes over keys, wave softmax, lanes over dims.
// ---------------------------------------------------------------------------
__global__ __launch_bounds__(32)
void dilated_attn_kernel(const float* __restrict__ qkv, float* __restrict__ o)
{
    const int idx  = blockIdx.x;
    const int q    = idx & (LL - 1);
    const int h    = (idx >> 11) & (HH - 1);
    const int b    = idx >> 14;
    const int lane = threadIdx.x;

    __shared__ float qs[HD];
    __shared__ float probs[WIN / DIL + 4];

    const size_t rowQ = (size_t)(b * LL + q) * (3 * DD);
    const float* qptr = qkv + rowQ + h * HD;
    qs[lane]      = qptr[lane];
    qs[lane + 32] = qptr[lane + 32];
    __syncthreads();

    const int nk = min(WIN / DIL, q >> 1) + 1;

    float s[5];
    float lmax = -3.0e38f;
    #pragma unroll
    for (int t = 0; t < 5; ++t) {
        const int j = lane + 32 * t;
        float sc = -3.0e38f;
        if (j < nk) {
            const int lk = q - DIL * j;
            const float* kptr = qkv + (size_t)(b * LL + lk) * (3 * DD) + DD + h * HD;
            float dot = 0.0f;
            #pragma unroll 8
            for (int d = 0; d < HD; ++d) dot += qs[d] * kptr[d];
            sc = dot * 0.125f;                   // 1/sqrt(64)
            lmax = fmaxf(lmax, sc);
        }
        s[t] = sc;
    }
    #pragma unroll
    for (int m = 16; m >= 1; m >>= 1) lmax = fmaxf(lmax, __shfl_xor(lmax, m, 32));

    float lsum = 0.0f;
    #pragma unroll
    for (int t = 0; t < 5; ++t) {
        const int j = lane + 32 * t;
        if (j < nk) { const float e = __expf(s[t] - lmax); s[t] = e; lsum += e; }
    }
    #pragma unroll
    for (int m = 16; m >= 1; m >>= 1) lsum += __shfl_xor(lsum, m, 32);
    const float inv = 1.0f / lsum;

    #pragma unroll
    for (int t = 0; t < 5; ++t) {
        const int j = lane + 32 * t;
        if (j < nk) probs[j] = s[t] * inv;
    }
    __syncthreads();

    const int d0 = lane << 1;
    float a0 = 0.0f, a1 = 0.0f;
    for (int j = 0; j < nk; ++j) {
        const int lk = q - DIL * j;
        const float p = probs[j];
        const float* vptr = qkv + (size_t)(b * LL + lk) * (3 * DD) + 2 * DD + h * HD + d0;
        a0 += p * vptr[0];
        a1 += p * vptr[1];
    }
    float* op = o + (size_t)(b * LL + q) * DD + h * HD + d0;
    op[0] = a0;
    op[1] = a1;
}

// ---------------------------------------------------------------------------
// Launch: LN1 -> QKV gemm -> attention -> out-proj(+res) -> LN2
//         -> FFN1(+GELU) -> FFN2(+res) -> d_out
// ---------------------------------------------------------------------------
extern "C" void kernel_launch(void* const* d_in, const int* in_sizes, int n_in,
                              void* d_out, int out_size, void* d_ws, size_t ws_size,
                              hipStream_t stream)
{
    const float* x       = (const float*)d_in[0];
    const float* norm1_w = (const float*)d_in[1];
    const float* norm1_b = (const float*)d_in[2];
    const float* qkv_w   = (const float*)d_in[3];
    const float* qkv_b   = (const float*)d_in[4];
    const float* out_w   = (const float*)d_in[5];
    const float* out_b   = (const float*)d_in[6];
    const float* norm2_w = (const float*)d_in[7];
    const float* norm2_b = (const float*)d_in[8];
    const float* ffn_w1  = (const float*)d_in[9];
    const float* ffn_b1  = (const float*)d_in[10];
    const float* ffn_w2  = (const float*)d_in[11];
    const float* ffn_b2  = (const float*)d_in[12];
    float* out = (float*)d_out;

    // Workspace layout (floats)
    float* ws = (float*)d_ws;
    const size_t SZ_TOKD = (size_t)NTOK * DD;        // 2M floats
    float* h    = ws;                                 // LN output (reused)
    float* qkv  = h    + SZ_TOKD;                     // [NTOK, 3D]
    float* attn = qkv  + (size_t)NTOK * 3 * DD;       // [NTOK, D]
    float* x1   = attn + SZ_TOKD;                     // residual-1
    float* f1   = x1   + SZ_TOKD;                     // [NTOK, HIDDEN]

    const dim3 blk128(128, 1, 1);
    const dim3 blk32(32, 1, 1);

    // 1) LN1: x -> h
    ln_kernel<<<NTOK, blk128, 0, stream>>>(x, norm1_w, norm1_b, h);

    // 2) QKV projection: qkv = h @ qkv_w^T + qkv_b   [4096 x 1536]
    gemm_wmma_kernel<<<dim3((3 * DD) / 64, NTOK / 64), blk128, 0, stream>>>(
        h, qkv_w, qkv_b, nullptr, qkv, NTOK, 3 * DD, DD, 0);

    // 3) Sparse dilated attention: qkv -> attn  [B,L,D]
    dilated_attn_kernel<<<BB * HH * LL, blk32, 0, stream>>>(qkv, attn);

    // 4) Output projection + residual: x1 = attn @ out_w^T + out_b + x
    gemm_wmma_kernel<<<dim3(DD / 64, NTOK / 64), blk128, 0, stream>>>(
        attn, out_w, out_b, x, x1, NTOK, DD, DD, 0);

    // 5) LN2: x1 -> h
    ln_kernel<<<NTOK, blk128, 0, stream>>>(x1, norm2_w, norm2_b, h);

    // 6) FFN up + exact GELU: f1 = gelu(h @ ffn_w1^T + ffn_b1)  [4096 x 2048]
    gemm_wmma_kernel<<<dim3(HIDDEN / 64, NTOK / 64), blk128, 0, stream>>>(
        h, ffn_w1, ffn_b1, nullptr, f1, NTOK, HIDDEN, DD, 1);

    // 7) FFN down + residual: out = f1 @ ffn_w2^T + ffn_b2 + x1
    gemm_wmma_kernel<<<dim3(DD / 64, NTOK / 64), blk128, 0, stream>>>(
        f1, ffn_w2, ffn_b2, x1, out, NTOK, DD, HIDDEN, 0);
}